// EdgeNetwork_81647328297533
// MI455X (gfx1250) — compile-verified
//
#include <hip/hip_runtime.h>
#include <hip/hip_bf16.h>

#define B_ 2
#define N_ 1024
#define D_ 16
#define H_ 8

typedef float v2f __attribute__((ext_vector_type(2)));
typedef float v8f __attribute__((ext_vector_type(8)));

// ---------------------------------------------------------------------------
// tanh: prefer the gfx1250 native V_TANH_F32 (single trans-pipe op; listed in
// the CDNA5 S_DELAY_ALU transcendental set). Fallback: branch-free
// exp2/rcp formulation (7 flat VALU ops, no EXEC divergence).
// ---------------------------------------------------------------------------
__device__ __forceinline__ float fast_tanh(float x) {
#if __has_builtin(__builtin_amdgcn_tanhf)
  return __builtin_amdgcn_tanhf(x);
#else
  float ax = __builtin_fabsf(x);
  float e  = __builtin_amdgcn_exp2f(ax * -2.885390081777927f); // -2*log2(e)
  float r  = (1.0f - e) * __builtin_amdgcn_rcpf(1.0f + e);
  return __builtin_copysignf(r, x);
#endif
}

// ---------------------------------------------------------------------------
// Prologue: per (b,n) compute t_j[b,n,h] and u[b,n,h] = t_i + t_v + b1.
// 2048 threads total; trivially cheap.
// ---------------------------------------------------------------------------
__global__ void edge_pre_kernel(const float* __restrict__ x,
                                const float* __restrict__ vp,
                                const float* __restrict__ W1,
                                const float* __restrict__ b1,
                                float* __restrict__ tj,
                                float* __restrict__ u) {
  int idx = blockIdx.x * blockDim.x + threadIdx.x;   // b*N + n
  if (idx >= B_ * N_) return;
  int b = idx >> 10;

  float xr[16], vr[16];
#pragma unroll
  for (int c = 0; c < 16; ++c) xr[c] = x[idx * 16 + c];
#pragma unroll
  for (int c = 0; c < 16; ++c) vr[c] = vp[b * 16 + c];

#pragma unroll
  for (int h = 0; h < 8; ++h) {
    const float* w = W1 + h * 49;          // [j(16) | i(16) | v(16) | A(1)]
    float sj = 0.f, si = 0.f, sv = 0.f;
#pragma unroll
    for (int c = 0; c < 16; ++c) {
      sj = fmaf(xr[c], w[c], sj);
      si = fmaf(xr[c], w[16 + c], si);
      sv = fmaf(vr[c], w[32 + c], sv);
    }
    tj[idx * 8 + h] = sj;
    u[idx * 8 + h]  = si + sv + b1[h];
  }
}

// ---------------------------------------------------------------------------
// A-matrix fragment for V_WMMA_F32_16X16X4_F32:
//   VGPR0: lanes0-15 = W[m, hbase+0], lanes16-31 = W[m, hbase+2]
//   VGPR1: lanes0-15 = W[m, hbase+1], lanes16-31 = W[m, hbase+3]
// Rows m>=8 zero-padded (H=8).
// ---------------------------------------------------------------------------
__device__ __forceinline__ v2f load_wfrag(const float* __restrict__ W,
                                          int m, int hi, int hbase) {
  v2f a;
  if (m < 8) {
    a.x = W[m * 8 + hbase + 2 * hi];
    a.y = W[m * 8 + hbase + 2 * hi + 1];
  } else {
    a.x = 0.f;
    a.y = 0.f;
  }
  return a;
}

// ---------------------------------------------------------------------------
// One MLP layer over a 16-pair tile: d[v] (lanes0-15: h[v][p]) -> tanh(W@h + b)
// B fragment built from previous D layout with 4 shfl_xor(16) + selects.
// Two chained f32 WMMAs cover K = 0..3 and 4..7.
// ---------------------------------------------------------------------------
__device__ __forceinline__ void mlp_layer(float d[8], v2f a0, v2f a1,
                                          const float* __restrict__ bias,
                                          int hi) {
  float u2 = __shfl_xor(d[2], 16, 32);
  float u3 = __shfl_xor(d[3], 16, 32);
  float u6 = __shfl_xor(d[6], 16, 32);
  float u7 = __shfl_xor(d[7], 16, 32);

  v2f b0, b1v;
  b0.x  = hi ? u2 : d[0];   // K=0 | K=2 rows of B (4x16)
  b0.y  = hi ? u3 : d[1];   // K=1 | K=3
  b1v.x = hi ? u6 : d[4];   // K=4 | K=6
  b1v.y = hi ? u7 : d[5];   // K=5 | K=7

  v8f c;
#pragma unroll
  for (int v = 0; v < 8; ++v) c[v] = bias[v];   // init accumulator with bias

  c = __builtin_amdgcn_wmma_f32_16x16x4_f32(false, a0, false, b0,
                                            (short)0, c, false, false);
  c = __builtin_amdgcn_wmma_f32_16x16x4_f32(false, a1, false, b1v,
                                            (short)0, c, false, false);
#pragma unroll
  for (int v = 0; v < 8; ++v) d[v] = fast_tanh(c[v]);
}

// ---------------------------------------------------------------------------
// Main kernel: one block per (b,i); 8 waves each loop over 8 tiles of 16 j's.
// ---------------------------------------------------------------------------
__global__ __launch_bounds__(256) void edge_main_kernel(
    const float* __restrict__ A,  const float* __restrict__ W1,
    const float* __restrict__ W2, const float* __restrict__ b2,
    const float* __restrict__ W3, const float* __restrict__ b3,
    const float* __restrict__ W4, const float* __restrict__ b4,
    const float* __restrict__ W5, const float* __restrict__ b5,
    const float* __restrict__ tj, const float* __restrict__ u,
    float* __restrict__ out) {
  const int lane = threadIdx.x & 31;
  const int wave = threadIdx.x >> 5;
  const int m    = lane & 15;
  const int hi   = lane >> 4;
  const int bi   = blockIdx.x;          // b*N + i
  const int b    = bi >> 10;

  // Per-wave constant weight fragments (reused for all 64 tiles of this row).
  v2f a2_0 = load_wfrag(W2, m, hi, 0), a2_1 = load_wfrag(W2, m, hi, 4);
  v2f a3_0 = load_wfrag(W3, m, hi, 0), a3_1 = load_wfrag(W3, m, hi, 4);
  v2f a4_0 = load_wfrag(W4, m, hi, 0), a4_1 = load_wfrag(W4, m, hi, 4);

  float w1a[8], ui[8], w5r[8];
#pragma unroll
  for (int h = 0; h < 8; ++h) {
    w1a[h] = W1[h * 49 + 48];                 // uniform -> s_load
    ui[h]  = u[(size_t)bi * 8 + h];           // uniform per block
    w5r[h] = W5[h];
  }
  const float bb5 = b5[0];

  const float* Arow = A  + (size_t)bi * N_;
  const float* tjb  = tj + (size_t)b * N_ * 8;
  float*       orow = out + (size_t)bi * N_;

#pragma unroll 2
  for (int t = wave; t < N_ / 16; t += 8) {
    const int    j   = t * 16 + m;            // lanes16-31 mirror lanes0-15
    const float  av  = __builtin_nontemporal_load(Arow + j);  // streamed once
    const float* tjr = tjb + (size_t)j * 8;

    // Layer 1: d[v] = tanh(t_j + u_i + A*W1_A) in D layout (lanes0-15 valid).
    float d[8];
#pragma unroll
    for (int h = 0; h < 8; ++h)
      d[h] = fast_tanh(fmaf(av, w1a[h], tjr[h] + ui[h]));

    mlp_layer(d, a2_0, a2_1, b2, hi);
    mlp_layer(d, a3_0, a3_1, b3, hi);
    mlp_layer(d, a4_0, a4_1, b4, hi);

    // Final: out[p] = b5 + sum_k W5[k] * h4[k][p]; column already per-lane.
    float acc = bb5;
#pragma unroll
    for (int v = 0; v < 8; ++v) acc = fmaf(w5r[v], d[v], acc);

    if (lane < 16) __builtin_nontemporal_store(acc, orow + t * 16 + m);
  }
}

// ---------------------------------------------------------------------------
extern "C" void kernel_launch(void* const* d_in, const int* in_sizes, int n_in,
                              void* d_out, int out_size, void* d_ws,
                              size_t ws_size, hipStream_t stream) {
  const float* x  = (const float*)d_in[0];
  const float* A  = (const float*)d_in[1];
  const float* vp = (const float*)d_in[2];
  const float* W1 = (const float*)d_in[3];
  const float* b1 = (const float*)d_in[4];
  const float* W2 = (const float*)d_in[5];
  const float* b2 = (const float*)d_in[6];
  const float* W3 = (const float*)d_in[7];
  const float* b3 = (const float*)d_in[8];
  const float* W4 = (const float*)d_in[9];
  const float* b4 = (const float*)d_in[10];
  const float* W5 = (const float*)d_in[11];
  const float* b5 = (const float*)d_in[12];

  float* tj = (float*)d_ws;                  // B*N*H floats = 64 KB
  float* u  = tj + (size_t)B_ * N_ * H_;     // another 64 KB

  edge_pre_kernel<<<(B_ * N_ + 255) / 256, 256, 0, stream>>>(x, vp, W1, b1, tj, u);
  edge_main_kernel<<<B_ * N_, 256, 0, stream>>>(A, W1, W2, b2, W3, b3, W4, b4,
                                                W5, b5, tj, u, (float*)d_out);
}